// DepthConsistencyLoss_24163486008062
// MI455X (gfx1250) — compile-verified
//
#include <hip/hip_runtime.h>

// Problem constants (match reference)
#define HH 256
#define WW 256
#define BB 8
#define NN 500000
#define HWW (HH * WW)

// Scatter config: 40 blocks per batch, each with a private 256KB LDS image
// (CDNA5 WGP has 320KB LDS -> one such workgroup per WGP).
#define BPB 40
#define NPC (NN / BPB)               // 12500 points per chunk (exact)
#define NG4 (NPC / 4)                // 3125 groups of 4 points per chunk (exact)
#define SCATTER_THREADS 1024

// Reduction config: 512*256 threads == BB*HWW/4 -> exactly one float4 per
// thread, no loop, EXEC all-1s at the WMMA (ISA requirement).
#define RED_BLOCKS 512
#define RED_THREADS 256
#define N_WAVES (RED_BLOCKS * (RED_THREADS / 32))   // 4096 per-wave partials

typedef __attribute__((ext_vector_type(2))) float v2f;
typedef __attribute__((ext_vector_type(8))) float v8f;

// ds_swizzle_b32 group-of-32 xor fold: offset = (xor_mask<<10) | and_mask(0x1f)
#define SWZ_XOR_ADD(v, m) \
    ((v) += __int_as_float(__builtin_amdgcn_ds_swizzle(__float_as_int(v), ((m) << 10) | 0x1F)))

__global__ void DCL_zero_kernel(float* __restrict__ ws, int n) {
    int i = blockIdx.x * blockDim.x + threadIdx.x;
    if (i < n) ws[i] = 0.0f;
}

__device__ __forceinline__ void DCL_proj_one(float x, float y, float z, float d,
                                             float* __restrict__ img) {
    if (d > 0.5f) {
        // (FX*x/z + CX)*W with FX=1, CX=0.5, W=256; int cast truncates like astype(int32)
        int u = (int)((x / z + 0.5f) * 256.0f);
        int v = (int)((y / z + 0.5f) * 256.0f);
        u = min(max(u, 0), WW - 1);
        v = min(max(v, 0), HH - 1);
        atomicAdd(&img[v * WW + u], z * d);   // ds_add_f32
    }
}

// Phase 1: project points, accumulate z*d into a per-workgroup private LDS
// image via ds_add_f32, then flush non-zero pixels with global_atomic_add_f32.
// All global reads are 16B-aligned b128: 4 points = 3 float4 + 1 float4 dens.
__global__ __launch_bounds__(SCATTER_THREADS)
void DCL_scatter_kernel(const float* __restrict__ points,
                        const float* __restrict__ dens,
                        float* __restrict__ acc) {
    extern __shared__ float img[];   // HWW floats = 256KB dynamic LDS
    const int b     = blockIdx.x / BPB;
    const int chunk = blockIdx.x % BPB;

    for (int i = threadIdx.x; i < HWW; i += SCATTER_THREADS) img[i] = 0.0f;
    __syncthreads();

    const int start = chunk * NPC;
    // chunk bases: start*3*4B = chunk*150000B and start*4B = chunk*50000B, both %16==0
    const float4* __restrict__ p4 =
        (const float4*)(points + (size_t)b * NN * 3 + (size_t)start * 3);
    const float4* __restrict__ d4 =
        (const float4*)(dens + (size_t)b * NN + start);

    for (int g = threadIdx.x; g < NG4; g += SCATTER_THREADS) {
        float4 pa = p4[3 * g + 0];   // x0 y0 z0 x1
        float4 pb = p4[3 * g + 1];   // y1 z1 x2 y2
        float4 pc = p4[3 * g + 2];   // z2 x3 y3 z3
        float4 dd = d4[g];
        DCL_proj_one(pa.x, pa.y, pa.z, dd.x, img);
        DCL_proj_one(pa.w, pb.x, pb.y, dd.y, img);
        DCL_proj_one(pb.z, pb.w, pc.x, dd.z, img);
        DCL_proj_one(pc.y, pc.z, pc.w, dd.w, img);
    }
    __syncthreads();

    float* __restrict__ accb = acc + (size_t)b * HWW;
    for (int i = threadIdx.x; i < HWW; i += SCATTER_THREADS) {
        float v = img[i];
        if (v != 0.0f) atomicAdd(&accb[i], v);    // global_atomic_add_f32
    }
}

// Phase 2: masked diff^2 + count, one float4 pair per thread (b128 loads),
// wave-folded through the matrix pipe. A = all-ones 16x4 => D[m][n] =
// sum_k B[k][n] (all rows identical); each lane fills one unique B slot with
// its partial (second slot 0), so folding row 0's 16 columns yields the
// 32-lane total. Per-wave results go to workspace (no atomics).
__global__ __launch_bounds__(RED_THREADS)
void DCL_reduce_kernel(const float* __restrict__ acc,
                       const float* __restrict__ depth,
                       float* __restrict__ psum,
                       float* __restrict__ pcnt) {
    const int i4 = blockIdx.x * RED_THREADS + (int)threadIdx.x;   // [0, BB*HWW/4)
    float4 p = ((const float4*)acc)[i4];
    float4 dp = ((const float4*)depth)[i4];

    float ls = 0.0f, lc = 0.0f;
    {   // branch-free: v_cmp + v_cndmask, EXEC untouched
        float m0 = (p.x > 0.0f) ? 1.0f : 0.0f;
        float m1 = (p.y > 0.0f) ? 1.0f : 0.0f;
        float m2 = (p.z > 0.0f) ? 1.0f : 0.0f;
        float m3 = (p.w > 0.0f) ? 1.0f : 0.0f;
        float e0 = p.x - dp.x, e1 = p.y - dp.y, e2 = p.z - dp.z, e3 = p.w - dp.w;
        ls = m0 * e0 * e0 + m1 * e1 * e1 + m2 * e2 * e2 + m3 * e3 * e3;
        lc = m0 + m1 + m2 + m3;
    }

    v2f a  = {1.0f, 1.0f};
    v2f bs = {ls, 0.0f};
    v2f bc = {lc, 0.0f};
    v8f cs = {};
    v8f cc = {};
    // 8 args: (neg_a, A, neg_b, B, c_mod, C, reuse_a, reuse_b)
    cs = __builtin_amdgcn_wmma_f32_16x16x4_f32(false, a, false, bs, (short)0, cs, false, false);
    cc = __builtin_amdgcn_wmma_f32_16x16x4_f32(false, a, false, bc, (short)0, cc, false, false);

    // C/D layout: VGPR0 = {M=0,N=lane} for lanes 0..15 (identical row M=8 for
    // lanes 16..31). Fold 16 columns with 4 ds_swizzle xor steps (no guards).
    float s = cs[0];
    float c = cc[0];
    SWZ_XOR_ADD(s, 1); SWZ_XOR_ADD(s, 2); SWZ_XOR_ADD(s, 4); SWZ_XOR_ADD(s, 8);
    SWZ_XOR_ADD(c, 1); SWZ_XOR_ADD(c, 2); SWZ_XOR_ADD(c, 4); SWZ_XOR_ADD(c, 8);

    if ((threadIdx.x & 31) == 0) {
        const int wid = blockIdx.x * (RED_THREADS / 32) + ((int)threadIdx.x >> 5);
        psum[wid] = s;
        pcnt[wid] = c;
    }
}

// Phase 3: single wave folds the 4096 per-wave partials and divides.
__global__ __launch_bounds__(32)
void DCL_finalize_kernel(const float* __restrict__ psum,
                         const float* __restrict__ pcnt,
                         float* __restrict__ out) {
    float s = 0.0f, c = 0.0f;
    for (int i = (int)threadIdx.x; i < N_WAVES; i += 32) {
        s += psum[i];
        c += pcnt[i];
    }
    SWZ_XOR_ADD(s, 1); SWZ_XOR_ADD(s, 2); SWZ_XOR_ADD(s, 4); SWZ_XOR_ADD(s, 8); SWZ_XOR_ADD(s, 16);
    SWZ_XOR_ADD(c, 1); SWZ_XOR_ADD(c, 2); SWZ_XOR_ADD(c, 4); SWZ_XOR_ADD(c, 8); SWZ_XOR_ADD(c, 16);
    if (threadIdx.x == 0) {
        if (c < 1.0f) c = 1.0f;   // max(count, 1)
        out[0] = s / c;
    }
}

extern "C" void kernel_launch(void* const* d_in, const int* in_sizes, int n_in,
                              void* d_out, int out_size, void* d_ws, size_t ws_size,
                              hipStream_t stream) {
    const float* points = (const float*)d_in[0];   // (B, N, 3) f32
    const float* dens   = (const float*)d_in[1];   // (B, N, 1) f32
    const float* depth  = (const float*)d_in[2];   // (B, 1, H, W) f32
    float* out = (float*)d_out;                    // scalar f32

    float* acc  = (float*)d_ws;                    // B*H*W accumulator
    float* psum = acc + (size_t)BB * HWW;          // N_WAVES per-wave diff2 sums
    float* pcnt = psum + N_WAVES;                  // N_WAVES per-wave counts

    // Zero only the accumulator; partials are fully overwritten every call.
    const int nzero = BB * HWW;
    DCL_zero_kernel<<<(nzero + 255) / 256, 256, 0, stream>>>(acc, nzero);

    DCL_scatter_kernel<<<BB * BPB, SCATTER_THREADS, HWW * sizeof(float), stream>>>(
        points, dens, acc);

    DCL_reduce_kernel<<<RED_BLOCKS, RED_THREADS, 0, stream>>>(acc, depth, psum, pcnt);

    DCL_finalize_kernel<<<1, 32, 0, stream>>>(psum, pcnt, out);
}